// pyGAMNet_18210661335092
// MI455X (gfx1250) — compile-verified
//
#include <hip/hip_runtime.h>

typedef __attribute__((ext_vector_type(2))) float v2f;
typedef __attribute__((ext_vector_type(8))) float v8f;

#define B_SAMPLES 131072
#define NFEAT 48
#define NCAT 16
#define BLOCKS_PER_FEAT 128
#define MT_PER_WAVE 8   // 8 waves * 8 mtiles * 16 samples = 1024 samples/block; *128 blocks = 131072

// Merge two partial-sum arrays across xor distance k: lanes with 'bit' clear end
// up tracking 'a', lanes with 'bit' set track 'b'. One shfl instead of two.
__device__ __forceinline__ float merge_reduce(float a, float b, int k, bool bit) {
    float keep = bit ? b : a;
    float send = bit ? a : b;
    return keep + __shfl_xor(send, k, 32);
}

// ---------------------------------------------------------------------------
// Numeric-feature kernel: per (feature, 16-sample tile), compute
//   h0 = relu(x*W0+b0)        [VALU, built directly in WMMA A-layout]
//   h1 = relu(h0 @ W1 + b1)   [V_WMMA_F32_16X16X4_F32, K=40 -> 10 steps, N=20 -> 2 tiles]
//   y  = h1 @ W2 + b2         [per-lane scale + merging-butterfly reduction]
// ---------------------------------------------------------------------------
__global__ __launch_bounds__(256)
void gam_num_kernel(const float* __restrict__ inputs,
                    const float* __restrict__ W0, const float* __restrict__ b0,
                    const float* __restrict__ W1, const float* __restrict__ b1,
                    const float* __restrict__ W2, const float* __restrict__ b2,
                    float* __restrict__ out)
{
    __shared__ float lw0[40];
    __shared__ float lb0[40];
    __shared__ float lw1[800];
    __shared__ float lb1[20];
    __shared__ float lw2[20];
    __shared__ float lb2[1];

    const int feat = blockIdx.x / BLOCKS_PER_FEAT;
    const int blk  = blockIdx.x % BLOCKS_PER_FEAT;
    const int tid  = threadIdx.x;

    // Stage this feature's weights into LDS (3.7 KB)
    if (tid < 40) { lw0[tid] = W0[feat * 40 + tid]; lb0[tid] = b0[feat * 40 + tid]; }
    for (int i = tid; i < 800; i += 256) lw1[i] = W1[feat * 800 + i];
    if (tid < 20) { lb1[tid] = b1[feat * 20 + tid]; lw2[tid] = W2[feat * 20 + tid]; }
    if (tid == 0) lb2[0] = b2[feat];
    __syncthreads();

    const int lane = tid & 31;
    const int wave = tid >> 5;
    const int half = lane >> 4;   // 0: lanes 0-15, 1: lanes 16-31
    const int lo   = lane & 15;

    // --- W1 as WMMA B-matrix tiles (4x16 per k-step), resident across M-tiles.
    // B 4x16 layout: VGPR r holds row (4kk + r) for lanes 0-15 and row (4kk + 2 + r)
    // for lanes 16-31, column = lane & 15 (N-tile 1 covers N=16..19, rest padded 0).
    v2f bm[2][10];
    #pragma unroll
    for (int nt = 0; nt < 2; ++nt) {
        const int  ncol  = nt * 16 + lo;
        const bool valid = (ncol < 20);
        #pragma unroll
        for (int kk = 0; kk < 10; ++kk) {
            const int kr = 4 * kk + 2 * half;
            float e0 = valid ? lw1[kr * 20 + ncol]       : 0.0f;
            float e1 = valid ? lw1[(kr + 1) * 20 + ncol] : 0.0f;
            bm[nt][kk] = (v2f){e0, e1};
        }
    }

    // --- Layer-0 weights/biases resident in registers (per-lane K slice).
    v2f w0r[10], b0r[10];
    #pragma unroll
    for (int kk = 0; kk < 10; ++kk) {
        const int k0 = 4 * kk + 2 * half;
        w0r[kk] = (v2f){lw0[k0], lw0[k0 + 1]};
        b0r[kk] = (v2f){lb0[k0], lb0[k0 + 1]};
    }

    // Per-lane layer-2 coefficients (lane owns column N = lo of the C tile).
    // Padded columns (N >= 20) have zero B, zero bias, zero W2 => contribute 0.
    const float w2a = lw2[lo];
    const float b1a = lb1[lo];
    const float w2b = (lo < 4) ? lw2[16 + lo] : 0.0f;
    const float b1b = (lo < 4) ? lb1[16 + lo] : 0.0f;
    const float bias2 = lb2[0];

    // Prefetch all sample values this wave needs (hides scattered-load latency
    // behind the WMMA chains; addresses are known upfront).
    const int mbase = blk * (MT_PER_WAVE * 8 * 16);
    float xs[MT_PER_WAVE];
    #pragma unroll
    for (int t = 0; t < MT_PER_WAVE; ++t) {
        const int m0 = mbase + (t * 8 + wave) * 16;
        xs[t] = inputs[(size_t)(m0 + lo) * 64 + feat];
    }

    #pragma unroll
    for (int t = 0; t < MT_PER_WAVE; ++t) {
        const int m0 = mbase + (t * 8 + wave) * 16;
        const float x = xs[t];

        // Layer 0: each lane builds its slice of the A-matrix (16x40 activations).
        // A 16x4 layout: lanes 0-15 hold M=lane, K=4kk+{0,1}; lanes 16-31 K=4kk+{2,3}.
        v2f am[10];
        #pragma unroll
        for (int kk = 0; kk < 10; ++kk) {
            float h0 = fmaxf(fmaf(x, w0r[kk][0], b0r[kk][0]), 0.0f);
            float h1 = fmaxf(fmaf(x, w0r[kk][1], b0r[kk][1]), 0.0f);
            am[kk] = (v2f){h0, h1};
        }

        // Layer 1: seed accumulators with the layer-1 bias (C layout: every M row
        // in a lane shares column N = lo), then 10 K-steps x 2 N-tiles of fp32
        // WMMA, interleaved as two independent accumulation chains for ILP.
        v8f c0 = {b1a, b1a, b1a, b1a, b1a, b1a, b1a, b1a};
        v8f c1 = {b1b, b1b, b1b, b1b, b1b, b1b, b1b, b1b};
        #pragma unroll
        for (int kk = 0; kk < 10; ++kk) {
            c0 = __builtin_amdgcn_wmma_f32_16x16x4_f32(
                false, am[kk], false, bm[0][kk], (short)0, c0, false, false);
            c1 = __builtin_amdgcn_wmma_f32_16x16x4_f32(
                false, am[kk], false, bm[1][kk], (short)0, c1, false, false);
        }

        // Layer 2: relu, scale by W2, then a merging butterfly reduction over the
        // 16 lanes of each half: 8 row-arrays collapse 8->4->2->1 with lane-parity
        // selects, so only 8 shfls total. Afterwards lane lo (lo<8 valid twice)
        // holds the full sum for row M = (lo&7) + 8*half.
        float p[8];
        #pragma unroll
        for (int v = 0; v < 8; ++v) {
            float h1a = fmaxf(c0[v], 0.0f);
            float h1b = fmaxf(c1[v], 0.0f);      // exactly 0 for lanes lo >= 4
            p[v] = fmaf(h1a, w2a, h1b * w2b);
        }
        float q0 = merge_reduce(p[0], p[1], 1, (lo & 1) != 0);
        float q1 = merge_reduce(p[2], p[3], 1, (lo & 1) != 0);
        float q2 = merge_reduce(p[4], p[5], 1, (lo & 1) != 0);
        float q3 = merge_reduce(p[6], p[7], 1, (lo & 1) != 0);
        float r0 = merge_reduce(q0, q1, 2, (lo & 2) != 0);
        float r1 = merge_reduce(q2, q3, 2, (lo & 2) != 0);
        float s  = merge_reduce(r0, r1, 4, (lo & 4) != 0);
        s += __shfl_xor(s, 8, 32);

        // Single predicated store: lanes 0-7 write rows m0..m0+7,
        // lanes 16-23 write rows m0+8..m0+15.
        if ((lo & 8) == 0)
            out[(size_t)(m0 + (lo & 7) + 8 * half) * 64 + feat] = s + bias2;
    }
}

// ---------------------------------------------------------------------------
// Categorical kernel: out[:, 48+c] = class_bias[c, (int)inputs[:, 48+c]]
// ---------------------------------------------------------------------------
__global__ __launch_bounds__(256)
void gam_cat_kernel(const float* __restrict__ inputs,
                    const float* __restrict__ class_bias,
                    float* __restrict__ out)
{
    const int gid = blockIdx.x * 256 + threadIdx.x;
    if (gid >= B_SAMPLES * NCAT) return;
    const int m = gid >> 4;
    const int c = gid & 15;
    const int col = 48 + c;
    const int idx = (int)inputs[(size_t)m * 64 + col];
    out[(size_t)m * 64 + col] = class_bias[c * 32 + idx];
}

extern "C" void kernel_launch(void* const* d_in, const int* in_sizes, int n_in,
                              void* d_out, int out_size, void* d_ws, size_t ws_size,
                              hipStream_t stream)
{
    const float* inputs     = (const float*)d_in[0];
    const float* W0         = (const float*)d_in[1];
    const float* b0         = (const float*)d_in[2];
    const float* W1         = (const float*)d_in[3];
    const float* b1         = (const float*)d_in[4];
    const float* W2         = (const float*)d_in[5];
    const float* b2         = (const float*)d_in[6];
    const float* class_bias = (const float*)d_in[7];
    float* out = (float*)d_out;

    gam_num_kernel<<<NFEAT * BLOCKS_PER_FEAT, 256, 0, stream>>>(
        inputs, W0, b0, W1, b1, W2, b2, out);

    gam_cat_kernel<<<(B_SAMPLES * NCAT) / 256, 256, 0, stream>>>(
        inputs, class_bias, out);
}